// GCN_VI_simple_58248346468474
// MI455X (gfx1250) — compile-verified
//
#include <hip/hip_runtime.h>
#include <math.h>

typedef __attribute__((ext_vector_type(2))) float v2f;
typedef __attribute__((ext_vector_type(8))) float v8f;

// ---------------------------------------------------------------------------
// Kernel 1: h[i] = dot(x[i,:], W[:])  via V_WMMA_F32_16X16X4_F32.
// One wave32 per 16-row tile. B-fragment carries W in column 0 only, so
// D column 0 accumulates the f32 GEMV exactly (full f32 precision).
// ---------------------------------------------------------------------------
__global__ __launch_bounds__(256) void gcn_gemv_wmma_f32(
    const float* __restrict__ x, const float* __restrict__ W,
    float* __restrict__ h, int N, int C) {
  const int lane = threadIdx.x & 31;
  const int waveInBlk = threadIdx.x >> 5;
  const int tile = blockIdx.x * (blockDim.x >> 5) + waveInBlk;
  const int row0 = tile << 4;
  if (row0 >= N) return;  // wave-uniform exit: EXEC stays all-ones below

  const int m    = lane & 15;        // row within tile (A) / col (D)
  const int koff = (lane >> 4) << 1; // lanes 16-31 hold K+2,K+3
  int row = row0 + m;
  if (row >= N) row = N - 1;         // clamp loads for ragged tail (none here)
  const float* __restrict__ xrow = x + (size_t)row * (size_t)C;
  const float bmask = (m == 0) ? 1.0f : 0.0f;  // only B column N=0 is live

  v8f acc = {};
  for (int k = 0; k < C; k += 4) {
    // A fragment: 8B per lane, rows reuse cachelines across k iterations (L0)
    v2f a = *(const v2f*)(xrow + k + koff);
    // B fragment: W[k+koff .. +1] into (K,N=0); branch-free mask keeps EXEC=~0
    v2f w = *(const v2f*)(W + k + koff);
    v2f bf;
    bf.x = w.x * bmask;
    bf.y = w.y * bmask;
    // 8 args: (neg_a, A, neg_b, B, c_mod, C, reuse_a, reuse_b)
    acc = __builtin_amdgcn_wmma_f32_16x16x4_f32(
        false, a, false, bf, (short)0, acc, false, false);
  }

  // D column 0: lane 0 holds M=0..7 in VGPR 0..7, lane 16 holds M=8..15
  if (m == 0) {
    const int mbase = row0 + ((lane >> 4) << 3);
#pragma unroll
    for (int r = 0; r < 8; ++r) {
      const int rr = mbase + r;
      if (rr < N) h[rr] = acc[r];
    }
  }
}

// ---------------------------------------------------------------------------
// Kernel 2: deg[i] = 1  (self-loop contribution)
// ---------------------------------------------------------------------------
__global__ void gcn_deg_init(float* __restrict__ deg, int N) {
  const int i = blockIdx.x * blockDim.x + threadIdx.x;
  if (i < N) deg[i] = 1.0f;
}

// ---------------------------------------------------------------------------
// Kernel 3: deg[col[e]] += 1 for every edge
// ---------------------------------------------------------------------------
__global__ void gcn_deg_count(const long long* __restrict__ col,
                              float* __restrict__ deg, int E) {
  const int i = blockIdx.x * blockDim.x + threadIdx.x;
  if (i < E) atomicAdd(&deg[(int)col[i]], 1.0f);
}

// ---------------------------------------------------------------------------
// Kernel 4: dinv = rsqrt(deg); acc seeded with self-loop term h[i]*dinv[i]^2
// ---------------------------------------------------------------------------
__global__ void gcn_dinv_seed(const float* __restrict__ deg,
                              const float* __restrict__ h,
                              float* __restrict__ dinv,
                              float* __restrict__ acc, int N) {
  const int i = blockIdx.x * blockDim.x + threadIdx.x;
  if (i < N) {
    const float d  = deg[i];              // always >= 1 (self-loops)
    const float di = rsqrtf(d);
    dinv[i] = di;
    acc[i]  = h[i] * di * di;
  }
}

// ---------------------------------------------------------------------------
// Kernel 5: acc[col] += h[row] * dinv[row] * dinv[col]   (gathers hit L2)
// ---------------------------------------------------------------------------
__global__ void gcn_scatter(const long long* __restrict__ row,
                            const long long* __restrict__ col,
                            const float* __restrict__ h,
                            const float* __restrict__ dinv,
                            float* __restrict__ acc, int E) {
  const int i = blockIdx.x * blockDim.x + threadIdx.x;
  if (i < E) {
    const int r = (int)row[i];
    const int c = (int)col[i];
    atomicAdd(&acc[c], h[r] * dinv[r] * dinv[c]);
  }
}

// ---------------------------------------------------------------------------
// Kernel 6: out = sigmoid(acc + b)
// ---------------------------------------------------------------------------
__global__ void gcn_finalize(const float* __restrict__ acc,
                             const float* __restrict__ b,
                             float* __restrict__ out, int N) {
  const int i = blockIdx.x * blockDim.x + threadIdx.x;
  if (i < N) {
    const float v = acc[i] + b[0];
    out[i] = 1.0f / (1.0f + expf(-v));
  }
}

// ---------------------------------------------------------------------------
// Host launcher
// ---------------------------------------------------------------------------
extern "C" void kernel_launch(void* const* d_in, const int* in_sizes, int n_in,
                              void* d_out, int out_size, void* d_ws, size_t ws_size,
                              hipStream_t stream) {
  const float*     x  = (const float*)d_in[0];       // [N, C]
  const long long* ei = (const long long*)d_in[1];   // [2, E] int64
  const float*     W  = (const float*)d_in[2];       // [1, C]
  const float*     b  = (const float*)d_in[3];       // [1]

  const int C = in_sizes[2];            // F_OUT(=1) * C_FEAT
  const int N = in_sizes[0] / C;        // 100000
  const int E = (int)(in_sizes[1] / 2); // 3200000

  const long long* rowIdx = ei;         // edge_index[0] (source)
  const long long* colIdx = ei + E;     // edge_index[1] (target)

  float* h    = (float*)d_ws;           // [N]
  float* deg  = h + N;                  // [N]
  float* dinv = deg + N;                // [N]
  float* acc  = dinv + N;               // [N]
  float* out  = (float*)d_out;          // [N]

  // 1) GEMV via WMMA: one wave per 16 rows, 8 waves (256 threads) per block
  {
    const int tiles = (N + 15) / 16;
    const int wavesPerBlk = 8;
    const int blocks = (tiles + wavesPerBlk - 1) / wavesPerBlk;
    gcn_gemv_wmma_f32<<<blocks, wavesPerBlk * 32, 0, stream>>>(x, W, h, N, C);
  }

  const int TB = 256;
  const int gN = (N + TB - 1) / TB;
  const int gE = (E + TB - 1) / TB;

  // 2) deg = 1 (self-loops)
  gcn_deg_init<<<gN, TB, 0, stream>>>(deg, N);
  // 3) deg[col] += 1
  gcn_deg_count<<<gE, TB, 0, stream>>>(colIdx, deg, E);
  // 4) dinv = rsqrt(deg); acc = h * dinv^2 (self-loop message)
  gcn_dinv_seed<<<gN, TB, 0, stream>>>(deg, h, dinv, acc, N);
  // 5) acc[col] += h[row] * dinv[row] * dinv[col]
  gcn_scatter<<<gE, TB, 0, stream>>>(rowIdx, colIdx, h, dinv, acc, E);
  // 6) out = sigmoid(acc + b)
  gcn_finalize<<<gN, TB, 0, stream>>>(acc, b, out, N);
}